// GAT_53137335386212
// MI455X (gfx1250) — compile-verified
//
#include <hip/hip_runtime.h>
#include <hip/hip_bf16.h>

#define N_NODES 50000
#define N_EDGES 800000
#define DIM 64

typedef float v2f __attribute__((ext_vector_type(2)));
typedef float v8f __attribute__((ext_vector_type(8)));

// ---------------------------------------------------------------------------
// GEMM: z = h @ W^T and zi = h @ U^T via V_WMMA_F32_16X16X4_F32.
// Grid: 3125 blocks (16 rows each), 4 waves/block (one 16-col tile per wave).
// A (16x4 f32, 2 VGPR): lanes 0-15 hold K=0(v0),K=1(v1); lanes 16-31 K=2,K=3.
// B (4x16 f32, 2 VGPR): lanes 0-15 hold rows K=0(v0),K=1(v1) cols N=lane;
//                       lanes 16-31 rows K=2,K=3 cols N=lane-16.
// C/D (16x16 f32, 8 VGPR): vgpr r = row r (lanes 0-15) / row r+8 (lanes 16-31).
// ---------------------------------------------------------------------------
__global__ __launch_bounds__(128)
void gat_gemm_wmma(const float* __restrict__ h,
                   const float* __restrict__ W,
                   const float* __restrict__ U,
                   float* __restrict__ z,
                   float* __restrict__ zi)
{
    const int lane = threadIdx.x & 31;
    const int wv   = threadIdx.x >> 5;        // 0..3 : output column tile
    const int i0   = blockIdx.x * 16;         // output row tile
    const int j0   = wv * 16;

    const int hi   = lane >> 4;               // 0 = lanes 0-15, 1 = lanes 16-31
    const int l15  = lane & 15;
    const int arow = i0 + l15;                // A-matrix row owned by this lane
    const int koff = hi * 2;                  // K sub-offset within 4-wide step

    v8f accW = {};
    v8f accU = {};
#pragma unroll
    for (int kk = 0; kk < 16; ++kk) {
        const int k = kk * 4 + koff;
        v2f a, bw, bu;
        a.x  = h[arow * DIM + k];
        a.y  = h[arow * DIM + k + 1];
        // B = W^T tile: B[k][j] = W[j][k]
        bw.x = W[(j0 + l15) * DIM + k];
        bw.y = W[(j0 + l15) * DIM + k + 1];
        bu.x = U[(j0 + l15) * DIM + k];
        bu.y = U[(j0 + l15) * DIM + k + 1];
        accW = __builtin_amdgcn_wmma_f32_16x16x4_f32(false, a, false, bw,
                                                     (short)0, accW, false, false);
        accU = __builtin_amdgcn_wmma_f32_16x16x4_f32(false, a, false, bu,
                                                     (short)0, accU, false, false);
    }
#pragma unroll
    for (int r = 0; r < 8; ++r) {
        const int orow = i0 + r + hi * 8;
        z [orow * DIM + j0 + l15] = accW[r];
        zi[orow * DIM + j0 + l15] = accU[r];
    }
}

// ---------------------------------------------------------------------------
// Per-node attention scores: ssrc[n] = z[n] . A[0:64], sdst[n] = z[n] . A[64:128]
// One wave32 per node, 2 elements per lane, shuffle reduction.
// ---------------------------------------------------------------------------
__global__ __launch_bounds__(256)
void gat_scores(const float* __restrict__ z, const float* __restrict__ A,
                float* __restrict__ ssrc, float* __restrict__ sdst, int n)
{
    const int lane = threadIdx.x & 31;
    const int node = (int)((blockIdx.x * blockDim.x + threadIdx.x) >> 5);
    if (node >= n) return;
    const float z0 = z[node * DIM + lane];
    const float z1 = z[node * DIM + 32 + lane];
    float ps = z0 * A[lane]      + z1 * A[lane + 32];
    float pd = z0 * A[64 + lane] + z1 * A[96 + lane];
#pragma unroll
    for (int off = 16; off > 0; off >>= 1) {
        ps += __shfl_down(ps, off, 32);
        pd += __shfl_down(pd, off, 32);
    }
    if (lane == 0) { ssrc[node] = ps; sdst[node] = pd; }
}

// ---------------------------------------------------------------------------
// Init per-layer accumulators: m = -inf, denom = deg = 0, agg = 0.
// ---------------------------------------------------------------------------
__global__ __launch_bounds__(256)
void gat_init(float* __restrict__ m, float* __restrict__ denom,
              float* __restrict__ deg, float* __restrict__ agg, int n)
{
    const int i = blockIdx.x * blockDim.x + threadIdx.x;
    if (i < n * DIM) agg[i] = 0.0f;
    if (i < n) { m[i] = -__builtin_inff(); denom[i] = 0.0f; deg[i] = 0.0f; }
}

// ---------------------------------------------------------------------------
// Edge logits: e = leaky_relu(ssrc[src] + sdst[dst] + d*V00*aE); segment max
// via float atomicMax (GLOBAL_ATOMIC_MAX_NUM_F32) and degree count.
// ---------------------------------------------------------------------------
__global__ __launch_bounds__(256)
void gat_edge_logits(const float* __restrict__ ssrc, const float* __restrict__ sdst,
                     const float* __restrict__ dfeat, const int* __restrict__ src,
                     const int* __restrict__ dst, const float* __restrict__ V,
                     const float* __restrict__ A, float* __restrict__ ebuf,
                     float* __restrict__ m, float* __restrict__ deg, int ne)
{
    const int e = blockIdx.x * blockDim.x + threadIdx.x;
    if (e >= ne) return;
    const int s = src[e];
    const int t = dst[e];
    float val = ssrc[s] + sdst[t] + dfeat[e] * V[0] * A[2 * DIM];
    val = (val > 0.0f) ? val : 0.01f * val;   // leaky_relu slope 0.01
    ebuf[e] = val;
    atomicMax(&m[t], val);
    atomicAdd(&deg[t], 1.0f);
}

// ---------------------------------------------------------------------------
// ex = exp(e - m[dst]); denom[dst] += ex.
// ---------------------------------------------------------------------------
__global__ __launch_bounds__(256)
void gat_edge_exp(const float* __restrict__ ebuf, const int* __restrict__ dst,
                  const float* __restrict__ m, float* __restrict__ ex,
                  float* __restrict__ denom, int ne)
{
    const int e = blockIdx.x * blockDim.x + threadIdx.x;
    if (e >= ne) return;
    const int t = dst[e];
    const float v = expf(ebuf[e] - m[t]);
    ex[e] = v;
    atomicAdd(&denom[t], v);
}

// ---------------------------------------------------------------------------
// agg[dst] += alpha * z[src]; one wave32 per edge, 2 features per lane.
// Gathers/scatters stay inside the 192MB L2 (z is 12.8MB).
// ---------------------------------------------------------------------------
__global__ __launch_bounds__(256)
void gat_aggregate(const float* __restrict__ ex, const float* __restrict__ denom,
                   const int* __restrict__ src, const int* __restrict__ dst,
                   const float* __restrict__ z, float* __restrict__ agg, int ne)
{
    const int widx = (int)((blockIdx.x * blockDim.x + threadIdx.x) >> 5);
    const int lane = threadIdx.x & 31;
    if (widx >= ne) return;
    const int s = src[widx];
    const int t = dst[widx];
    const float alpha = ex[widx] / denom[t];
    atomicAdd(&agg[t * DIM + lane],      alpha * z[s * DIM + lane]);
    atomicAdd(&agg[t * DIM + 32 + lane], alpha * z[s * DIM + 32 + lane]);
}

// ---------------------------------------------------------------------------
// out = deg > 0 ? relu(zi + agg) : 0
// ---------------------------------------------------------------------------
__global__ __launch_bounds__(256)
void gat_finalize(const float* __restrict__ zi, const float* __restrict__ agg,
                  const float* __restrict__ deg, float* __restrict__ out, int n)
{
    const int i = blockIdx.x * blockDim.x + threadIdx.x;
    if (i >= n * DIM) return;
    const int node = i >> 6;
    float v = zi[i] + agg[i];
    v = (v > 0.0f) ? v : 0.0f;
    out[i] = (deg[node] > 0.0f) ? v : 0.0f;
}

// ---------------------------------------------------------------------------
// Host-side orchestration (one GAT layer)
// ---------------------------------------------------------------------------
static void run_gat_layer(const float* h, const float* dfeat,
                          const int* src, const int* dst,
                          const float* V, const float* W, const float* U,
                          const float* A,
                          float* z, float* zi, float* agg,
                          float* ssrc, float* sdst, float* m,
                          float* denom, float* deg, float* ex,
                          float* out, hipStream_t stream)
{
    const int n = N_NODES, ne = N_EDGES;
    // 50000 = 3125 * 16 exactly -> no tail, EXEC all-ones for WMMA
    gat_gemm_wmma<<<n / 16, 128, 0, stream>>>(h, W, U, z, zi);
    gat_scores<<<(n * 32 + 255) / 256, 256, 0, stream>>>(z, A, ssrc, sdst, n);
    gat_init<<<(n * DIM + 255) / 256, 256, 0, stream>>>(m, denom, deg, agg, n);
    gat_edge_logits<<<(ne + 255) / 256, 256, 0, stream>>>(ssrc, sdst, dfeat, src, dst,
                                                          V, A, ex /*ebuf*/, m, deg, ne);
    gat_edge_exp<<<(ne + 255) / 256, 256, 0, stream>>>(ex, dst, m, ex, denom, ne);
    gat_aggregate<<<(ne * 32 + 255) / 256, 256, 0, stream>>>(ex, denom, src, dst,
                                                             z, agg, ne);
    gat_finalize<<<(n * DIM + 255) / 256, 256, 0, stream>>>(zi, agg, deg, out, n);
}

extern "C" void kernel_launch(void* const* d_in, const int* in_sizes, int n_in,
                              void* d_out, int out_size, void* d_ws, size_t ws_size,
                              hipStream_t stream)
{
    const float* h    = (const float*)d_in[0];
    const float* df   = (const float*)d_in[1];
    const int*   src  = (const int*)  d_in[2];
    const int*   dst  = (const int*)  d_in[3];
    const float* V0   = (const float*)d_in[4];
    const float* W0   = (const float*)d_in[5];
    const float* U0   = (const float*)d_in[6];
    const float* A0   = (const float*)d_in[7];
    const float* V1   = (const float*)d_in[8];
    const float* W1   = (const float*)d_in[9];
    const float* U1   = (const float*)d_in[10];
    const float* A1   = (const float*)d_in[11];
    float* out = (float*)d_out;

    // Workspace layout (floats): 4*N*DIM + 5*N + E  ~= 55.4 MB
    float* ws    = (float*)d_ws;
    float* z     = ws;
    float* zi    = z    + (size_t)N_NODES * DIM;
    float* agg   = zi   + (size_t)N_NODES * DIM;
    float* h1    = agg  + (size_t)N_NODES * DIM;
    float* ssrc  = h1   + (size_t)N_NODES * DIM;
    float* sdst  = ssrc + N_NODES;
    float* m     = sdst + N_NODES;
    float* denom = m    + N_NODES;
    float* deg   = denom+ N_NODES;
    float* ex    = deg  + N_NODES;

    // Layer 1: h -> h1 (workspace)
    run_gat_layer(h,  df, src, dst, V0, W0, U0, A0,
                  z, zi, agg, ssrc, sdst, m, denom, deg, ex, h1, stream);
    // Layer 2: h1 -> d_out
    run_gat_layer(h1, df, src, dst, V1, W1, U1, A1,
                  z, zi, agg, ssrc, sdst, m, denom, deg, ex, out, stream);
}